// VQCEncoder_45938970198546
// MI455X (gfx1250) — compile-verified
//
#include <hip/hip_runtime.h>
#include <math.h>

typedef __attribute__((ext_vector_type(2))) float v2f;
typedef __attribute__((ext_vector_type(8))) float v8f;

#define FDIM 512
#define WSTRIDE 516   // 516 mod 64 == 4 -> disjoint bank pairs for the two half-waves
#define CSTRIDE 20    // epilogue tile pad

__device__ __forceinline__ float fast_tanh(float x) {
#if __has_builtin(__builtin_amdgcn_tanhf)
    return __builtin_amdgcn_tanhf(x);
#elif __has_builtin(__builtin_amdgcn_tanh_f32)
    return __builtin_amdgcn_tanh_f32(x);
#else
    // tanh(x) = 1 - 2/(exp(2x)+1): branch-free, saturates to +/-1 correctly
    float e = __expf(2.0f * x);
    return 1.0f - __fdividef(2.0f, e + 1.0f);
#endif
}

__global__ __launch_bounds__(256)
void vqc_encoder_kernel(const float* __restrict__ x,
                        const float* __restrict__ Wp,    // (8,512)
                        const float* __restrict__ bp,    // (8,)
                        const float* __restrict__ wv,    // (2,2,3) flat
                        float* __restrict__ out)         // (B,11)
{
    __shared__ float Wlds[16][WSTRIDE];          // rows 8..15 zeroed (N padding)
    __shared__ float ctile[8][16][CSTRIDE];
    __shared__ float gcs[12], gsn[12];           // cos/sin(theta/2), shared by all rows

    const int tid  = threadIdx.x;
    const int wave = tid >> 5;
    const int lane = tid & 31;
    const int n    = lane & 15;   // N index (also row index M for A)
    const int hi   = lane >> 4;   // half-wave: K subgroup

    // Precompute the 12 half-angle sin/cos once per block (same for every row).
    if (tid < 12) {
        float s, c;
        __sincosf(0.5f * wv[tid], &s, &c);
        gsn[tid] = s;
        gcs[tid] = c;
    }

    // Stage B = W^T into LDS as Wlds[n][k] = W_proj[n][k]; zero-pad n>=8.
    for (int i = tid; i < 16 * WSTRIDE; i += 256) {
        int r = i / WSTRIDE, k = i - r * WSTRIDE;
        float v = 0.0f;
        if (r < 8 && k < FDIM) v = Wp[r * FDIM + k];
        Wlds[r][k] = v;
    }
    __syncthreads();

    const int tileRow = blockIdx.x * 128 + wave * 16;        // first row of this wave's tile
    const float* xrow = x + (size_t)(tileRow + n) * FDIM + 2 * hi;

    // D[m][n] = sum_k x[m][k] * W[n][k], via 128 x V_WMMA_F32_16X16X4_F32
    v8f acc = {};
    for (int k = 0; k < FDIM; k += 16) {
        __builtin_prefetch(xrow + k + 128, 0, 0);            // global_prefetch_b8, 512B ahead
        #pragma unroll
        for (int k4 = 0; k4 < 16; k4 += 4) {
            v2f a = *reinterpret_cast<const v2f*>(xrow + k + k4);               // A[m][..]
            v2f b = *reinterpret_cast<const v2f*>(&Wlds[n][k + k4 + 2 * hi]);   // B[..][n]
            acc = __builtin_amdgcn_wmma_f32_16x16x4_f32(
                false, a, false, b, (short)0, acc, false, false);
        }
    }

    // Spill C fragment: lane L holds D[r + 8*hi][n] in acc[r]
    #pragma unroll
    for (int r = 0; r < 8; ++r)
        ctile[wave][r + 8 * hi][n] = acc[r];
    __syncthreads();

    // ---------------- Epilogue: one lane per row (lanes 0..15) ----------------
    if (lane < 16) {
        const int m   = lane;
        const int row = tileRow + m;

        // bias + tanh + L2 norm
        float p[8];
        float ss = 0.0f;
        #pragma unroll
        for (int j = 0; j < 8; ++j) {
            float t = fast_tanh(ctile[wave][m][j] + bp[j]);
            p[j] = t;
            ss += t * t;
        }
        const float inv = __fdividef(1.0f, fmaxf(sqrtf(ss), 1e-12f));

        // amplitude-embedded state (real init)
        float re[8], im[8];
        #pragma unroll
        for (int i = 0; i < 8; ++i) { re[i] = p[i] * inv; im[i] = 0.0f; }

        // 2 layers: Ry(q0..q2), Rz(q0..q2), CNOT(0,1),(1,2),(2,0); qubit q <-> bit (4>>q)
        #pragma unroll
        for (int l = 0; l < 2; ++l) {
            #pragma unroll
            for (int q = 0; q < 3; ++q) {               // Ry
                const float c = gcs[l * 6 + q];
                const float s = gsn[l * 6 + q];
                const int st = 4 >> q;
                #pragma unroll
                for (int i = 0; i < 8; ++i) {
                    if (!(i & st)) {
                        const int j = i | st;
                        float r0 = re[i], r1 = re[j];
                        re[i] = c * r0 - s * r1;  re[j] = s * r0 + c * r1;
                        float i0 = im[i], i1 = im[j];
                        im[i] = c * i0 - s * i1;  im[j] = s * i0 + c * i1;
                    }
                }
            }
            #pragma unroll
            for (int q = 0; q < 3; ++q) {               // Rz: diag(e^{-i t/2}, e^{+i t/2})
                const float c = gcs[l * 6 + 3 + q];
                const float s = gsn[l * 6 + 3 + q];
                const int st = 4 >> q;
                #pragma unroll
                for (int i = 0; i < 8; ++i) {
                    const float sp = (i & st) ? s : -s;
                    float r = re[i], q2 = im[i];
                    re[i] = r * c - q2 * sp;
                    im[i] = r * sp + q2 * c;
                }
            }
            // CNOTs: swap (control=1, t=0) <-> (control=1, t=1)
            const int ctrl[3] = {0, 1, 2};
            const int targ[3] = {1, 2, 0};
            #pragma unroll
            for (int g = 0; g < 3; ++g) {
                const int sc = 4 >> ctrl[g];
                const int st = 4 >> targ[g];
                #pragma unroll
                for (int i = 0; i < 8; ++i) {
                    if ((i & sc) && !(i & st)) {
                        const int j = i | st;
                        float tr = re[i]; re[i] = re[j]; re[j] = tr;
                        float ti = im[i]; im[i] = im[j]; im[j] = ti;
                    }
                }
            }
        }

        // outputs: projected (pre-norm tanh values) then <Z_q>
        float* o = out + (size_t)row * 11;
        #pragma unroll
        for (int j = 0; j < 8; ++j) o[j] = p[j];
        #pragma unroll
        for (int q = 0; q < 3; ++q) {
            const int st = 4 >> q;
            float z = 0.0f;
            #pragma unroll
            for (int i = 0; i < 8; ++i) {
                const float pi = re[i] * re[i] + im[i] * im[i];
                z += (i & st) ? -pi : pi;
            }
            o[8 + q] = z;
        }
    }
}

extern "C" void kernel_launch(void* const* d_in, const int* in_sizes, int n_in,
                              void* d_out, int out_size, void* d_ws, size_t ws_size,
                              hipStream_t stream) {
    const float* x  = (const float*)d_in[0];
    const float* Wp = (const float*)d_in[1];
    const float* bp = (const float*)d_in[2];
    const float* wv = (const float*)d_in[3];
    float* out = (float*)d_out;

    const int B = in_sizes[0] / FDIM;           // 262144
    dim3 grid((B + 127) / 128), block(256);
    hipLaunchKernelGGL(vqc_encoder_kernel, grid, block, 0, stream,
                       x, Wp, bp, wv, out);
}